// GAT_57157424775338
// MI455X (gfx1250) — compile-verified
//
#include <hip/hip_runtime.h>
#include <hip/hip_bf16.h>
#include <math.h>

typedef __attribute__((ext_vector_type(2))) float v2f;
typedef __attribute__((ext_vector_type(8))) float v8f;

#define FDIM 256
#define HEADS 4
#define NEG_SLOPE 0.2f

// Monotonic float -> u32 key so GLOBAL_ATOMIC_MAX_U32 implements float max.
__device__ __forceinline__ unsigned fkey(float f) {
  unsigned b = __float_as_uint(f);
  return (b & 0x80000000u) ? ~b : (b | 0x80000000u);
}
__device__ __forceinline__ float fdec(unsigned k) {
  unsigned b = (k & 0x80000000u) ? (k & 0x7FFFFFFFu) : ~k;
  return __uint_as_float(b);
}

// feat = h @ W   [N,256] x [256,256], fp32 WMMA 16x16x4.
// Block = 256 thr (8 waves). Block computes 16 rows x 256 cols; each wave a
// 16x32 strip (2 accumulators). A tile staged in LDS, shared by all waves.
__global__ void __launch_bounds__(256) gat_gemm(const float* __restrict__ h,
                                                const float* __restrict__ W,
                                                float* __restrict__ feat, int N) {
  __shared__ float As[16 * FDIM];
  const int rowbase = blockIdx.x * 16;
  {
    int t = threadIdx.x;          // 0..255
    int r = t >> 4;               // 0..15
    int c = (t & 15) * 16;        // 16 floats per thread
    bool ok = (rowbase + r) < N;
    const float* srcp = h + (size_t)(rowbase + r) * FDIM + c;
    float4 z = make_float4(0.f, 0.f, 0.f, 0.f);
#pragma unroll
    for (int j = 0; j < 4; ++j) {
      float4 v = ok ? *(const float4*)(srcp + j * 4) : z;
      *(float4*)&As[r * FDIM + c + j * 4] = v;
    }
  }
  __syncthreads();

  const int lane = threadIdx.x & 31;
  const int wave = threadIdx.x >> 5;
  const int colbase = wave * 32;
  const int mrow = lane & 15;
  const int kk = (lane >> 4) << 1;           // 0 (lanes 0-15) or 2 (lanes 16-31)
  const int ncol = colbase + (lane & 15);

  v8f acc0 = {};
  v8f acc1 = {};
  for (int k0 = 0; k0 < FDIM; k0 += 4) {
    // A frag: VGPR0 = K0/K2, VGPR1 = K1/K3 (contiguous pair in LDS row)
    v2f a;
    a.x = As[mrow * FDIM + k0 + kk];
    a.y = As[mrow * FDIM + k0 + kk + 1];
    const float* w0 = W + (size_t)(k0 + kk) * FDIM;
    const float* w1 = W + (size_t)(k0 + kk + 1) * FDIM;
    v2f b0, b1;
    b0.x = w0[ncol];      b0.y = w1[ncol];
    b1.x = w0[ncol + 16]; b1.y = w1[ncol + 16];
    acc0 = __builtin_amdgcn_wmma_f32_16x16x4_f32(false, a, false, b0, (short)0, acc0, false, false);
    acc1 = __builtin_amdgcn_wmma_f32_16x16x4_f32(false, a, false, b1, (short)0, acc1, false, false);
  }

  // C/D layout: VGPR g -> row = rowbase + g + (lane>=16 ? 8 : 0), col = colbase + lane%16
  const int rlo = rowbase + (lane >> 4) * 8;
  const int c0 = colbase + (lane & 15);
#pragma unroll
  for (int g = 0; g < 8; ++g) {
    int r = rlo + g;
    if (r < N) {
      feat[(size_t)r * FDIM + c0]      = acc0[g];
      feat[(size_t)r * FDIM + c0 + 16] = acc1[g];
    }
  }
}

// el[n,h] = sum_d feat[n,h,d]*al[h,d]; er likewise. One wave per node.
__global__ void __launch_bounds__(256) gat_attn_coef(const float* __restrict__ feat,
                                                     const float* __restrict__ al,
                                                     const float* __restrict__ ar,
                                                     float* __restrict__ el,
                                                     float* __restrict__ er, int N) {
  const int wave = threadIdx.x >> 5, lane = threadIdx.x & 31;
  const int node = blockIdx.x * 8 + wave;
  if (node >= N) return;
  const int c0 = lane * 8;  // 8 channels per lane; lanes 0-7 = head 0, etc.
  const float4* f = (const float4*)(feat + (size_t)node * FDIM + c0);
  const float4* A = (const float4*)(al + c0);
  const float4* B = (const float4*)(ar + c0);
  float4 f0 = f[0], f1 = f[1], a0 = A[0], a1 = A[1], b0 = B[0], b1 = B[1];
  float pl = f0.x * a0.x + f0.y * a0.y + f0.z * a0.z + f0.w * a0.w +
             f1.x * a1.x + f1.y * a1.y + f1.z * a1.z + f1.w * a1.w;
  float pr = f0.x * b0.x + f0.y * b0.y + f0.z * b0.z + f0.w * b0.w +
             f1.x * b1.x + f1.y * b1.y + f1.z * b1.z + f1.w * b1.w;
  for (int off = 1; off < 8; off <<= 1) {
    pl += __shfl_xor(pl, off, 32);
    pr += __shfl_xor(pr, off, 32);
  }
  if ((lane & 7) == 0) {
    el[node * HEADS + (lane >> 3)] = pl;
    er[node * HEADS + (lane >> 3)] = pr;
  }
}

// e = leaky_relu(el[src]+er[dst]); atomic segment-max into mkey[dst].
__global__ void __launch_bounds__(256) gat_edge_logits(const int* __restrict__ src,
                                                       const int* __restrict__ dst,
                                                       const float* __restrict__ el,
                                                       const float* __restrict__ er,
                                                       float* __restrict__ est,
                                                       unsigned* __restrict__ mkey, int EH) {
  int t = blockIdx.x * blockDim.x + threadIdx.x;
  if (t >= EH) return;
  int e = t >> 2, hh = t & 3;
  float v = el[src[e] * HEADS + hh] + er[dst[e] * HEADS + hh];
  v = v > 0.f ? v : NEG_SLOPE * v;
  est[t] = v;
  atomicMax(&mkey[dst[e] * HEADS + hh], fkey(v));
}

// ex = exp(e - m[dst]); den[dst] += ex.
__global__ void __launch_bounds__(256) gat_edge_exp(const int* __restrict__ dst,
                                                    const unsigned* __restrict__ mkey,
                                                    float* __restrict__ est,
                                                    float* __restrict__ den, int EH) {
  int t = blockIdx.x * blockDim.x + threadIdx.x;
  if (t >= EH) return;
  int e = t >> 2, hh = t & 3;
  int d = dst[e];
  float m = fdec(mkey[d * HEADS + hh]);
  if (!isfinite(m)) m = 0.f;
  float ex = __expf(est[t] - m);
  est[t] = ex;
  atomicAdd(&den[d * HEADS + hh], ex);
}

// out[dst] += (ex/den[dst]) * feat[src] ; 64 threads per edge, float4 each.
__global__ void __launch_bounds__(256) gat_edge_scatter(const int* __restrict__ src,
                                                        const int* __restrict__ dst,
                                                        const float* __restrict__ est,
                                                        const float* __restrict__ den,
                                                        const float* __restrict__ feat,
                                                        float* __restrict__ acc, int E) {
  long long t = (long long)blockIdx.x * blockDim.x + threadIdx.x;
  if (t >= (long long)E * 64) return;
  int e = (int)(t >> 6);
  int c4 = ((int)t & 63) * 4;
  int s = src[e], d = dst[e];
  int hh = c4 >> 6;  // head = channel/64
  float a = est[e * HEADS + hh] / fmaxf(den[d * HEADS + hh], 1e-9f);
  float4 f = *(const float4*)(feat + (size_t)s * FDIM + c4);
  float* o = acc + (size_t)d * FDIM + c4;
  atomicAdd(o + 0, a * f.x);
  atomicAdd(o + 1, a * f.y);
  atomicAdd(o + 2, a * f.z);
  atomicAdd(o + 3, a * f.w);
}

// out = relu(acc + residual) in place (res == nullptr -> no residual).
__global__ void __launch_bounds__(256) gat_finalize(float* __restrict__ acc,
                                                    const float* __restrict__ res, int N4) {
  int t = blockIdx.x * blockDim.x + threadIdx.x;
  if (t >= N4) return;
  float4 v = ((const float4*)acc)[t];
  if (res) {
    float4 r = ((const float4*)res)[t];
    v.x += r.x; v.y += r.y; v.z += r.z; v.w += r.w;
  }
  v.x = fmaxf(v.x, 0.f);
  v.y = fmaxf(v.y, 0.f);
  v.z = fmaxf(v.z, 0.f);
  v.w = fmaxf(v.w, 0.f);
  ((float4*)acc)[t] = v;
}

extern "C" void kernel_launch(void* const* d_in, const int* in_sizes, int n_in,
                              void* d_out, int out_size, void* d_ws, size_t ws_size,
                              hipStream_t stream) {
  const float* x   = (const float*)d_in[0];
  const int*   src = (const int*)d_in[1];
  const int*   dst = (const int*)d_in[2];
  const float* W[3]  = {(const float*)d_in[3], (const float*)d_in[6], (const float*)d_in[9]};
  const float* al[3] = {(const float*)d_in[4], (const float*)d_in[7], (const float*)d_in[10]};
  const float* ar[3] = {(const float*)d_in[5], (const float*)d_in[8], (const float*)d_in[11]};

  const int N = in_sizes[0] / FDIM;
  const int E = in_sizes[1];
  const int EH = E * HEADS;

  char* ws = (char*)d_ws;
  float* feat = (float*)ws;  ws += (size_t)N * FDIM * sizeof(float);
  float* bufA = (float*)ws;  ws += (size_t)N * FDIM * sizeof(float);
  float* bufB = (float*)ws;  ws += (size_t)N * FDIM * sizeof(float);
  float* el   = (float*)ws;  ws += (size_t)N * HEADS * sizeof(float);
  float* er   = (float*)ws;  ws += (size_t)N * HEADS * sizeof(float);
  float* den  = (float*)ws;  ws += (size_t)N * HEADS * sizeof(float);
  unsigned* mkey = (unsigned*)ws; ws += (size_t)N * HEADS * sizeof(unsigned);
  float* est  = (float*)ws;  ws += (size_t)EH * sizeof(float);

  float* outf = (float*)d_out;
  const float* lay_in[3] = {x, bufA, bufB};
  float*       lay_out[3] = {bufA, bufB, outf};

  for (int L = 0; L < 3; ++L) {
    hipMemsetAsync(lay_out[L], 0, (size_t)N * FDIM * sizeof(float), stream);
    hipMemsetAsync(den, 0, (size_t)N * HEADS * sizeof(float), stream);
    hipMemsetAsync(mkey, 0, (size_t)N * HEADS * sizeof(unsigned), stream);  // key 0 == -inf

    gat_gemm<<<(N + 15) / 16, 256, 0, stream>>>(lay_in[L], W[L], feat, N);
    gat_attn_coef<<<(N + 7) / 8, 256, 0, stream>>>(feat, al[L], ar[L], el, er, N);
    gat_edge_logits<<<(EH + 255) / 256, 256, 0, stream>>>(src, dst, el, er, est, mkey, EH);
    gat_edge_exp<<<(EH + 255) / 256, 256, 0, stream>>>(dst, mkey, est, den, EH);
    long long tot = (long long)E * 64;
    gat_edge_scatter<<<(int)((tot + 255) / 256), 256, 0, stream>>>(src, dst, est, den, feat,
                                                                   lay_out[L], E);
    int n4 = N * (FDIM / 4);
    gat_finalize<<<(n4 + 255) / 256, 256, 0, stream>>>(lay_out[L], L < 2 ? lay_in[L] : nullptr, n4);
  }
}